// LCAOConv_69423851373365
// MI455X (gfx1250) — compile-verified
//
#include <hip/hip_runtime.h>
#include <hip/hip_bf16.h>

#define NN   20000
#define EE   200000
#define TT   400000
#define HID  128
#define DOWN 32
#define CO   64
#define DR   16

typedef __attribute__((ext_vector_type(2))) float v2f;
typedef __attribute__((ext_vector_type(8))) float v8f;

__device__ __forceinline__ float silu_f(float v) {
    return v / (1.0f + __expf(-v));
}

// ---------------------------------------------------------------------------
// Zero-fill (float4 vectorized, grid-stride)
// ---------------------------------------------------------------------------
__global__ void zero_kernel(float4* __restrict__ p, long long n4) {
    long long i = (long long)blockIdx.x * blockDim.x + threadIdx.x;
    long long stride = (long long)gridDim.x * blockDim.x;
    float4 z; z.x = 0.f; z.y = 0.f; z.z = 0.f; z.w = 0.f;
    for (; i < n4; i += stride) p[i] = z;
}

// ---------------------------------------------------------------------------
// Two-layer MLP with WMMA f32 16x16x4.
//   out = act2( act1(silu(in) @ W1 (+b1)) @ W2 (+b2) )
//   act1 = silu, final layer linear (+optional bias).
// One wave (32 threads) per 16-row tile. A fragments staged via LDS.
// A-frag layout (16x4 f32):  lane L: M=L%16, v0=A[M][k+2*(L/16)], v1=A[M][k+2*(L/16)+1]
// B-frag layout (4x16 f32):  lane L: N=L%16, v0=B[k+2*(L/16)][N], v1=B[k+2*(L/16)+1][N]
// C/D layout: vgpr r: M=r+8*(L/16), N=L%16
// ---------------------------------------------------------------------------
template<int KIN, int KMID, int KOUT, bool BIAS>
__global__ __launch_bounds__(32) void mlp2_wmma(const float* __restrict__ in,
                                                const float* __restrict__ W1,
                                                const float* __restrict__ b1,
                                                const float* __restrict__ W2,
                                                const float* __restrict__ b2,
                                                float* __restrict__ out) {
    __shared__ float A[16 * (KIN + 1)];
    __shared__ float H[16 * (KMID + 1)];

    const int lane = threadIdx.x;
    const int row0 = blockIdx.x * 16;
    const int m  = lane & 15;
    const int hi = lane >> 4;

    __builtin_prefetch(W1, 0, 3);
    __builtin_prefetch(W2, 0, 3);

    // stage silu(in) tile into LDS
    for (int idx = lane; idx < 16 * KIN; idx += 32) {
        int r = idx / KIN, c0 = idx % KIN;
        float v = in[(size_t)(row0 + r) * KIN + c0];
        A[r * (KIN + 1) + c0] = silu_f(v);
    }
    __syncthreads();

    // GEMM1: [16,KIN] @ [KIN,KMID] -> silu -> H
    for (int nt = 0; nt < KMID / 16; ++nt) {
        const int n0 = nt * 16;
        v8f acc = {};
        for (int k = 0; k < KIN; k += 4) {
            v2f a, b;
            a[0] = A[m * (KIN + 1) + k + 2 * hi];
            a[1] = A[m * (KIN + 1) + k + 2 * hi + 1];
            b[0] = W1[(size_t)(k + 2 * hi) * KMID + n0 + m];
            b[1] = W1[(size_t)(k + 2 * hi + 1) * KMID + n0 + m];
            acc = __builtin_amdgcn_wmma_f32_16x16x4_f32(false, a, false, b,
                                                        (short)0, acc, false, false);
        }
        const int col = n0 + m;
        const float bias = BIAS ? b1[col] : 0.0f;
        for (int r = 0; r < 8; ++r) {
            float h = acc[r] + bias;
            H[(r + 8 * hi) * (KMID + 1) + col] = silu_f(h);
        }
    }
    __syncthreads();

    // GEMM2: [16,KMID] @ [KMID,KOUT] -> out
    for (int nt = 0; nt < KOUT / 16; ++nt) {
        const int n0 = nt * 16;
        v8f acc = {};
        for (int k = 0; k < KMID; k += 4) {
            v2f a, b;
            a[0] = H[m * (KMID + 1) + k + 2 * hi];
            a[1] = H[m * (KMID + 1) + k + 2 * hi + 1];
            b[0] = W2[(size_t)(k + 2 * hi) * KOUT + n0 + m];
            b[1] = W2[(size_t)(k + 2 * hi + 1) * KOUT + n0 + m];
            acc = __builtin_amdgcn_wmma_f32_16x16x4_f32(false, a, false, b,
                                                        (short)0, acc, false, false);
        }
        const int col = n0 + m;
        const float bias = BIAS ? b2[col] : 0.0f;
        for (int r = 0; r < 8; ++r) {
            out[(size_t)(row0 + r + 8 * hi) * KOUT + col] = acc[r] + bias;
        }
    }
}

// ---------------------------------------------------------------------------
// Final up-projection: out[N,128] = agg[N,32] @ W_up[32,128]  (WMMA, 1 wave/16 rows)
// ---------------------------------------------------------------------------
__global__ __launch_bounds__(32) void up_wmma(const float* __restrict__ agg,
                                              const float* __restrict__ W,
                                              float* __restrict__ out) {
    __shared__ float A[16 * 33];
    const int lane = threadIdx.x;
    const int row0 = blockIdx.x * 16;
    const int m  = lane & 15;
    const int hi = lane >> 4;

    for (int idx = lane; idx < 16 * 32; idx += 32) {
        int r = idx >> 5, c0 = idx & 31;
        A[r * 33 + c0] = agg[(size_t)(row0 + r) * DOWN + c0];
    }
    __syncthreads();

    for (int nt = 0; nt < HID / 16; ++nt) {
        const int n0 = nt * 16;
        v8f acc = {};
        for (int k = 0; k < DOWN; k += 4) {
            v2f a, b;
            a[0] = A[m * 33 + k + 2 * hi];
            a[1] = A[m * 33 + k + 2 * hi + 1];
            b[0] = W[(size_t)(k + 2 * hi) * HID + n0 + m];
            b[1] = W[(size_t)(k + 2 * hi + 1) * HID + n0 + m];
            acc = __builtin_amdgcn_wmma_f32_16x16x4_f32(false, a, false, b,
                                                        (short)0, acc, false, false);
        }
        for (int r = 0; r < 8; ++r) {
            out[(size_t)(row0 + r + 8 * hi) * HID + n0 + m] = acc[r];
        }
    }
}

// ---------------------------------------------------------------------------
// Triplet stage: one wave per t; lane = h channel (DOWN == 32 == wave32).
//   v = c[idx_k]*c[idx_j] + c[idx_k]; normalize over h per dr;
//   tw[edge_idx_ji] += (shbs*robs[edge_idx_kj]) * v_hat   (atomic segment sum)
// ---------------------------------------------------------------------------
__global__ __launch_bounds__(256) void triplet_kernel(const float* __restrict__ c,
                                                      const float* __restrict__ robs,
                                                      const float* __restrict__ shbs,
                                                      const int* __restrict__ idx_j,
                                                      const int* __restrict__ idx_k,
                                                      const int* __restrict__ e_kj,
                                                      const int* __restrict__ e_ji,
                                                      float* __restrict__ tw) {
    const int wid  = (int)((blockIdx.x * (long long)blockDim.x + threadIdx.x) >> 5);
    const int lane = threadIdx.x & 31;
    if (wid >= TT) return;

    const int ik  = idx_k[wid];
    const int ij  = idx_j[wid];
    const int ekj = e_kj[wid];
    const int eji = e_ji[wid];

    const float* ck  = c + (size_t)ik  * (DR * DOWN);
    const float* cj  = c + (size_t)ij  * (DR * DOWN);
    float*       twp = tw + (size_t)eji * (DR * DOWN);

    for (int dr = 0; dr < DR; ++dr) {
        float a = ck[dr * DOWN + lane];
        float b = cj[dr * DOWN + lane];
        float v = a * b + a;
        float ss = v * v;
        #pragma unroll
        for (int off = 16; off > 0; off >>= 1) ss += __shfl_xor(ss, off, 32);
        float inv = 1.0f / fmaxf(sqrtf(ss), 1e-12f);
        float s = shbs[(size_t)wid * DR + dr] * robs[(size_t)ekj * DR + dr];
        atomicAdd(&twp[dr * DOWN + lane], s * inv * v);
    }
}

// ---------------------------------------------------------------------------
// Edge stage: one wave per edge; lane = h channel.
//   u = (c[dst]*c[src] + c[dst]) * tw[e];  u_hat per dr;
//   r = normalize( sum_dr robs[e,dr]*u_hat );  agg[src] += x_d[dst]*r (atomic)
// ---------------------------------------------------------------------------
__global__ __launch_bounds__(256) void edge_kernel(const float* __restrict__ c,
                                                   const float* __restrict__ robs,
                                                   const float* __restrict__ tw,
                                                   const int* __restrict__ edge_index,
                                                   const float* __restrict__ x_d,
                                                   float* __restrict__ agg) {
    const int wid  = (int)((blockIdx.x * (long long)blockDim.x + threadIdx.x) >> 5);
    const int lane = threadIdx.x & 31;
    if (wid >= EE) return;

    const int src = edge_index[wid];            // edge_index[0][e]
    const int dst = edge_index[(size_t)EE + wid];  // edge_index[1][e]

    const float* cd  = c  + (size_t)dst * (DR * DOWN);
    const float* cs  = c  + (size_t)src * (DR * DOWN);
    const float* twe = tw + (size_t)wid * (DR * DOWN);

    float r = 0.0f;
    for (int dr = 0; dr < DR; ++dr) {
        float a = cd[dr * DOWN + lane];
        float b = cs[dr * DOWN + lane];
        float u = (a * b + a) * twe[dr * DOWN + lane];
        float ss = u * u;
        #pragma unroll
        for (int off = 16; off > 0; off >>= 1) ss += __shfl_xor(ss, off, 32);
        float inv = 1.0f / fmaxf(sqrtf(ss), 1e-12f);
        r += robs[(size_t)wid * DR + dr] * (u * inv);
    }
    float ss = r * r;
    #pragma unroll
    for (int off = 16; off > 0; off >>= 1) ss += __shfl_xor(ss, off, 32);
    float inv = 1.0f / fmaxf(sqrtf(ss), 1e-12f);
    r *= inv;

    float val = x_d[(size_t)dst * DOWN + lane] * r;
    atomicAdd(&agg[(size_t)src * DOWN + lane], val);
}

// ---------------------------------------------------------------------------
// Launch
// ---------------------------------------------------------------------------
extern "C" void kernel_launch(void* const* d_in, const int* in_sizes, int n_in,
                              void* d_out, int out_size, void* d_ws, size_t ws_size,
                              hipStream_t stream) {
    const float* x        = (const float*)d_in[0];
    const float* robs     = (const float*)d_in[1];
    const float* shbs     = (const float*)d_in[2];
    const float* coeffs   = (const float*)d_in[3];
    const int*   eidx     = (const int*)  d_in[4];
    const int*   idx_j    = (const int*)  d_in[5];
    const int*   idx_k    = (const int*)  d_in[6];
    const int*   e_kj     = (const int*)  d_in[7];
    const int*   e_ji     = (const int*)  d_in[8];
    const float* W_node1  = (const float*)d_in[9];
    const float* b_node1  = (const float*)d_in[10];
    const float* W_node2  = (const float*)d_in[11];
    const float* b_node2  = (const float*)d_in[12];
    const float* W_co1    = (const float*)d_in[13];
    const float* W_co2    = (const float*)d_in[14];
    const float* W_up     = (const float*)d_in[15];
    float* out = (float*)d_out;

    // workspace layout (floats)
    float* ws  = (float*)d_ws;
    float* x_d = ws;                                     //  N*DOWN        =    640,000
    float* agg = ws + (size_t)NN * DOWN;                 //  N*DOWN        =    640,000
    float* c   = ws + (size_t)2 * NN * DOWN;             //  N*DR*DOWN     = 10,240,000
    float* tw  = c  + (size_t)NN * DR * DOWN;            //  E*DR*DOWN     = 102,400,000

    // zero the atomic accumulation buffers
    zero_kernel<<<2048, 256, 0, stream>>>((float4*)agg, (long long)NN * DOWN / 4);
    zero_kernel<<<8192, 256, 0, stream>>>((float4*)tw, (long long)EE * DR * DOWN / 4);

    // x_d = silu(silu(x)@W1+b1)@W2+b2         [N,32]
    mlp2_wmma<HID, HID, DOWN, true><<<NN / 16, 32, 0, stream>>>(
        x, W_node1, b_node1, W_node2, b_node2, x_d);

    // c = silu(silu(coeffs)@W_co1)@W_co2      [N*DR,32]
    mlp2_wmma<CO, HID, DOWN, false><<<(NN * DR) / 16, 32, 0, stream>>>(
        coeffs, W_co1, nullptr, W_co2, nullptr, c);

    // triplet interaction -> tw               [E,DR,DOWN]
    triplet_kernel<<<(TT * 32) / 256, 256, 0, stream>>>(
        c, robs, shbs, idx_j, idx_k, e_kj, e_ji, tw);

    // edge interaction + aggregation -> agg   [N,DOWN]
    edge_kernel<<<(EE * 32) / 256, 256, 0, stream>>>(
        c, robs, tw, eidx, x_d, agg);

    // out = agg @ W_up                        [N,HID]
    up_wmma<<<NN / 16, 32, 0, stream>>>(agg, W_up, out);
}